// TemporalAttentionCausal_64458869179037
// MI455X (gfx1250) — compile-verified
//
#include <hip/hip_runtime.h>

typedef __attribute__((ext_vector_type(2))) float v2f;
typedef __attribute__((ext_vector_type(8))) float v8f;

#define B_DIM 8
#define L_DIM 16
#define T_DIM 1024
#define H_DIM 256
#define NTHREADS 256   // 8 wave32s

// One workgroup per (b,l). Pass 1: WMMA f32 16x16x4 matvec for scores.
// Pass 2: thread-per-channel sequential weighted prefix scan (coalesced).
__global__ __launch_bounds__(NTHREADS)
void temporal_attn_causal_kernel(const float* __restrict__ hin,
                                 const float* __restrict__ attn_w,
                                 const float* __restrict__ attn_b,
                                 float* __restrict__ out) {
    __shared__ float s_lds[T_DIM];   // scores, then exp(scores - max)
    __shared__ float w_lds[H_DIM];   // attention weight vector
    __shared__ float red[NTHREADS];  // max-reduction scratch

    const int tid  = threadIdx.x;
    const int lane = tid & 31;
    const int wv   = tid >> 5;       // wave id 0..7
    const int m16  = lane & 15;      // N index (t within tile) for B/D
    const int khi  = lane >> 4;      // 0 or 1: selects K pair {0,1} vs {2,3}

    const size_t bl = blockIdx.x;    // 0..127  == (b*L + l)
    const float* __restrict__ hb = hin + bl * (size_t)(T_DIM * H_DIM);
    float* __restrict__       ob = out + bl * (size_t)(T_DIM * H_DIM);

    // stage w into LDS (blockDim == H_DIM)
    w_lds[tid] = attn_w[tid];
    const float bias = attn_b[0];
    __syncthreads();

    // ---------------- Pass 1: scores via V_WMMA_F32_16X16X4_F32 ----------------
    // D[m,n] = sum_k A[m,k]*B[k,n];  A[m,k] = w[kb+k] (all rows equal),
    // B[k,n] = h[t0+n, kb+k]  ->  D[:,n] = s[t0+n] (pre-bias).
    // Layout (32-bit, 16x4 / 4x16): v0 holds K=0 (lanes 0-15) / K=2 (lanes 16-31),
    // v1 holds K=1 / K=3.  Lane's K base = 2*khi.
    for (int tile = 0; tile < 8; ++tile) {
        const int t0 = (wv * 8 + tile) * 16;
        const float* __restrict__ bp = hb + (size_t)(t0 + m16) * H_DIM + 2 * khi;
        v8f acc = {};
        #pragma unroll 8
        for (int kb = 0; kb < H_DIM; kb += 4) {
            v2f av, bv;
            av.x = w_lds[kb + 2 * khi];        // LDS broadcast (uniform per half-wave)
            av.y = w_lds[kb + 2 * khi + 1];
            bv.x = bp[kb];                     // h[t0+m16, kb+2*khi]
            bv.y = bp[kb + 1];                 // h[t0+m16, kb+2*khi+1]
            acc = __builtin_amdgcn_wmma_f32_16x16x4_f32(
                false, av, false, bv, (short)0, acc, false, false);
        }
        // every D row equals s[t0 + n]; lane n (0..15) publishes it
        if (lane < 16) s_lds[t0 + lane] = acc[0] + bias;
    }
    __syncthreads();

    // ---------------- Global max over T (matches reference) ----------------
    float m = -INFINITY;
    #pragma unroll
    for (int t = tid; t < T_DIM; t += NTHREADS) m = fmaxf(m, s_lds[t]);
    red[tid] = m;
    __syncthreads();
    for (int off = NTHREADS / 2; off > 0; off >>= 1) {
        if (tid < off) red[tid] = fmaxf(red[tid], red[tid + off]);
        __syncthreads();
    }
    const float smax = red[0];
    __syncthreads();

    // exp(s - smax) in place
    #pragma unroll
    for (int t = tid; t < T_DIM; t += NTHREADS) s_lds[t] = __expf(s_lds[t] - smax);
    __syncthreads();

    // ---------------- Pass 2: causal weighted prefix scan ----------------
    // Thread tid owns channel j = tid. Fully coalesced global loads/stores.
    // acc_e accumulated in the same sequential order as the reference cumsum.
    float acc_h = 0.0f;
    float acc_e = 0.0f;
    const float* __restrict__ hp = hb + tid;
    float* __restrict__       op = ob + tid;
    #pragma unroll 4
    for (int t = 0; t < T_DIM; ++t) {
        const float e  = s_lds[t];                 // LDS broadcast
        const float hv = hp[(size_t)t * H_DIM];    // L2-resident after pass 1
        acc_e += e;
        acc_h = fmaf(e, hv, acc_h);
        op[(size_t)t * H_DIM] = hv + acc_h / (acc_e + 1e-12f);
    }
}

extern "C" void kernel_launch(void* const* d_in, const int* in_sizes, int n_in,
                              void* d_out, int out_size, void* d_ws, size_t ws_size,
                              hipStream_t stream) {
    (void)in_sizes; (void)n_in; (void)out_size; (void)d_ws; (void)ws_size;
    const float* h  = (const float*)d_in[0];
    const float* w  = (const float*)d_in[1];
    const float* b  = (const float*)d_in[2];
    float* out = (float*)d_out;

    dim3 grid(B_DIM * L_DIM);   // 128 workgroups, one per (b,l)
    dim3 block(NTHREADS);       // 8 wave32s
    hipLaunchKernelGGL(temporal_attn_causal_kernel, grid, block, 0, stream,
                       h, w, b, out);
}